// GNN_47373489275402
// MI455X (gfx1250) — compile-verified
//
#include <hip/hip_runtime.h>
#include <cstdint>

typedef __attribute__((ext_vector_type(2))) float v2f;
typedef __attribute__((ext_vector_type(8))) float v8f;

#define NEG_SLOPE 0.01f

// ---------------------------------------------------------------------------
// Edge kernel: gather x[src] rows, atomic scatter-add into agg[tgt].
// 16 threads per edge; each thread moves one float4 (4 channels).
// Lanes 0-15 of a wave read one contiguous 256B row (perfectly coalesced per
// half-wave). Optionally accumulates float degree counts (layer 1 only).
// ---------------------------------------------------------------------------
__global__ __launch_bounds__(256)
void sage_scatter(const float* __restrict__ x,
                  const int*   __restrict__ src,
                  const int*   __restrict__ tgt,
                  float*       __restrict__ agg,
                  float*       __restrict__ deg,   // nullptr => skip
                  long long E)
{
    long long tid = (long long)blockIdx.x * blockDim.x + threadIdx.x;
    long long e   = tid >> 4;
    int       g   = (int)(tid & 15);
    if (e >= E) return;

    int s = src[e];
    int t = tgt[e];

    const float4 v = reinterpret_cast<const float4*>(x + (long long)s * 64)[g];
    float* dst = agg + (long long)t * 64 + g * 4;
    atomicAdd(dst + 0, v.x);
    atomicAdd(dst + 1, v.y);
    atomicAdd(dst + 2, v.z);
    atomicAdd(dst + 3, v.w);

    if (deg != nullptr && g == 0) {
        atomicAdd(deg + t, 1.0f);
    }
}

// ---------------------------------------------------------------------------
// Node kernel: out = leaky_relu( (agg/deg) @ Wl + b + xin @ Wr )
// One wave (32 lanes) per 16-row tile, full-precision V_WMMA_F32_16X16X4_F32.
// K = 64 (16 k-steps of 4). COUT in {64, 32} -> NT = COUT/16 column tiles with
// persistent v8f accumulators. Mean-normalization folded into the A fragment.
//
// f32 WMMA operand layouts (ISA 7.12.2):
//   A(16x4):  lane = M + 16*(K>>1), vgpr = K&1
//   B(4x16):  lane = N + 16*(K>>1), vgpr = K&1
//   C/D:      vgpr v, lane l -> M = v + 8*(l>>4), N = l&15
// ---------------------------------------------------------------------------
template <int COUT>
__global__ __launch_bounds__(32)
void sage_node(const float* __restrict__ agg,
               const float* __restrict__ xin,
               const float* __restrict__ Wl,    // [64 x COUT] row-major
               const float* __restrict__ Wr,    // [64 x COUT] row-major
               const float* __restrict__ bias,  // [COUT]
               const float* __restrict__ deg,   // [N]
               float*       __restrict__ out,   // [N x COUT]
               int N)
{
    constexpr int NT = COUT / 16;

    const int lane = threadIdx.x;          // 0..31
    const int m0   = blockIdx.x * 16;      // row-tile base
    const int half = lane >> 4;            // 0: K+0/1, 1: K+2/3
    const int col  = lane & 15;            // N within col tile
    int mrow = m0 + col;                   // A/deg row owned by this lane
    if (mrow >= N) mrow = N - 1;           // clamp (N is a multiple of 16 here)

    const float d     = deg[mrow];
    const float scale = d > 0.0f ? 1.0f / d : 0.0f;

    const float* __restrict__ arow = agg + (long long)mrow * 64;
    const float* __restrict__ xrow = xin + (long long)mrow * 64;

    v8f c[NT];
#pragma unroll
    for (int t = 0; t < NT; ++t)
        c[t] = (v8f){0.f, 0.f, 0.f, 0.f, 0.f, 0.f, 0.f, 0.f};

    // ---- pass 1: (agg * scale) @ Wl -------------------------------------
#pragma unroll
    for (int kk = 0; kk < 16; ++kk) {
        const int kb = 4 * kk + 2 * half;
        v2f a;
        a.x = arow[kb + 0] * scale;
        a.y = arow[kb + 1] * scale;
#pragma unroll
        for (int t = 0; t < NT; ++t) {
            v2f b;
            b.x = Wl[(long long)(kb + 0) * COUT + t * 16 + col];
            b.y = Wl[(long long)(kb + 1) * COUT + t * 16 + col];
            c[t] = __builtin_amdgcn_wmma_f32_16x16x4_f32(
                false, a, false, b, (short)0, c[t], false, false);
        }
    }

    // ---- pass 2: xin @ Wr ------------------------------------------------
#pragma unroll
    for (int kk = 0; kk < 16; ++kk) {
        const int kb = 4 * kk + 2 * half;
        v2f a;
        a.x = xrow[kb + 0];
        a.y = xrow[kb + 1];
#pragma unroll
        for (int t = 0; t < NT; ++t) {
            v2f b;
            b.x = Wr[(long long)(kb + 0) * COUT + t * 16 + col];
            b.y = Wr[(long long)(kb + 1) * COUT + t * 16 + col];
            c[t] = __builtin_amdgcn_wmma_f32_16x16x4_f32(
                false, a, false, b, (short)0, c[t], false, false);
        }
    }

    // ---- bias + leaky relu + store --------------------------------------
#pragma unroll
    for (int t = 0; t < NT; ++t) {
        const float bv = bias[t * 16 + col];
#pragma unroll
        for (int v = 0; v < 8; ++v) {
            const int row = m0 + v + 8 * half;
            float val = c[t][v] + bv;
            val = val > 0.0f ? val : NEG_SLOPE * val;
            if (row < N)
                out[(long long)row * COUT + t * 16 + col] = val;
        }
    }
}

// ---------------------------------------------------------------------------
// Host-side orchestration (graph-capture safe: only async ops on `stream`).
// setup_inputs order: x, edge_index, W1_l, b1, W1_r, W2_l, b2, W2_r
// ---------------------------------------------------------------------------
extern "C" void kernel_launch(void* const* d_in, const int* in_sizes, int n_in,
                              void* d_out, int out_size, void* d_ws, size_t ws_size,
                              hipStream_t stream)
{
    const float* x    = (const float*)d_in[0];
    const int*   ei   = (const int*)d_in[1];
    const float* W1_l = (const float*)d_in[2];
    const float* b1   = (const float*)d_in[3];
    const float* W1_r = (const float*)d_in[4];
    const float* W2_l = (const float*)d_in[5];
    const float* b2   = (const float*)d_in[6];
    const float* W2_r = (const float*)d_in[7];
    float*       outp = (float*)d_out;

    const long long E = in_sizes[1] / 2;     // edge_index is [2, E]
    const int       N = in_sizes[0] / 64;    // x is [N, 64]

    const int* src = ei;
    const int* tgt = ei + E;

    // Workspace layout (floats), 1KB-aligned chunks.
    float* ws   = (float*)d_ws;
    float* deg  = ws;                              // N
    float* agg1 = ws + 102400;                     // N*64
    float* h    = agg1 + (long long)N * 64;        // N*64
    float* agg2 = h    + (long long)N * 64;        // N*64
    (void)ws_size; (void)n_in; (void)out_size;

    // Zero accumulators (async memset is graph-capturable).
    hipMemsetAsync(deg,  0, (size_t)N * sizeof(float), stream);
    hipMemsetAsync(agg1, 0, (size_t)N * 64 * sizeof(float), stream);
    hipMemsetAsync(agg2, 0, (size_t)N * 64 * sizeof(float), stream);

    const long long scatter_threads = E * 16;
    const int scatter_blocks = (int)((scatter_threads + 255) / 256);
    const int node_blocks    = (N + 15) / 16;

    // Layer 1
    sage_scatter<<<scatter_blocks, 256, 0, stream>>>(x, src, tgt, agg1, deg, E);
    sage_node<64><<<node_blocks, 32, 0, stream>>>(agg1, x, W1_l, W1_r, b1, deg, h, N);

    // Layer 2 (reuses deg)
    sage_scatter<<<scatter_blocks, 256, 0, stream>>>(h, src, tgt, agg2, nullptr, E);
    sage_node<32><<<node_blocks, 32, 0, stream>>>(agg2, h, W2_l, W2_r, b2, deg, outp, N);
}